// UnifiedCognitiveLayer_71296457113952
// MI455X (gfx1250) — compile-verified
//
#include <hip/hip_runtime.h>
#include <math.h>

#ifndef __has_builtin
#define __has_builtin(x) 0
#endif

// Problem constants (match reference)
#define DIMK   2048
#define NSYM   1024
#define NBATCH 4
#define SEQ    4096
#define MTOT   (NBATCH * SEQ)   // 16384 tokens
#define KC     32               // K-chunk staged in LDS
#define NCHUNK (DIMK / KC)      // 64

typedef __attribute__((ext_vector_type(2))) float v2f;
typedef __attribute__((ext_vector_type(8))) float v8f;

// ---------------------------------------------------------------------------
// CDNA5 primitives
// ---------------------------------------------------------------------------
__device__ __forceinline__ v8f wmma_f32x4(v2f a, v2f b, v8f c) {
  return __builtin_amdgcn_wmma_f32_16x16x4_f32(false, a, false, b, (short)0, c,
                                               false, false);
}

// Async copy global -> LDS, 16 bytes, ASYNCcnt-tracked (GVS mode: 64-bit SGPR
// base + signed 32-bit VGPR byte offset).  ldsoff = low 32 bits of the generic
// shared address (LDS aperture maps addr[31:0] to the LDS byte address).
__device__ __forceinline__ void async_b128(unsigned ldsoff, unsigned goff,
                                           const void* sbase) {
  asm volatile("global_load_async_to_lds_b128 %0, %1, %2"
               :
               : "v"(ldsoff), "v"(goff), "s"(sbase)
               : "memory");
}

template <int N>
__device__ __forceinline__ void wait_async() {
#if __has_builtin(__builtin_amdgcn_s_wait_asynccnt)
  __builtin_amdgcn_s_wait_asynccnt((unsigned short)N);
#else
  asm volatile("s_wait_asynccnt %0" ::"i"(N) : "memory");
#endif
}

__device__ __forceinline__ unsigned lds_addr32(const void* p) {
  return (unsigned)(unsigned long long)p;  // low 32 bits = LDS byte address
}

// ---------------------------------------------------------------------------
// LDS tiles.  Rows padded so 16 consecutive-row fragment reads hit 16 banks:
//   f32: stride 36 floats (144 B, 16B-multiple for b128 copies).
//   u8 : stride 48 bytes.
// ---------------------------------------------------------------------------
typedef float TileF32[128][36];
typedef unsigned char TileU8[128][48];

// Stage 128 rows x 32 f32 from row-major [*, DIMK] matrix. 8 b128 per thread.
__device__ __forceinline__ void stage_f32(float (*dst)[36],
                                          const float* __restrict__ src,
                                          int rowBase, int k0, int t) {
  int r0 = t >> 3;          // 16 threads... 8 granules per 128B row
  int c0 = (t & 7) << 2;    // float column of this thread's 16B granule
  unsigned l0 = lds_addr32(&dst[r0][c0]);
  unsigned g0 =
      ((unsigned)(rowBase + r0) * (unsigned)DIMK + (unsigned)(k0 + c0)) * 4u;
#pragma unroll
  for (int i = 0; i < 8; ++i)  // +16 rows per step
    async_b128(l0 + 2304u * i /*16*144*/, g0 + 131072u * i /*16*DIMK*4*/, src);
}

// Stage 128 rows x 32 u8 spikes. 2 b128 per thread.
__device__ __forceinline__ void stage_u8(unsigned char (*dst)[48],
                                         const unsigned char* __restrict__ src,
                                         int rowBase, int k0, int t) {
  int r0 = t >> 1;
  int c0 = (t & 1) << 4;
  unsigned l0 = lds_addr32(&dst[r0][c0]);
  unsigned g0 = (unsigned)(rowBase + r0) * (unsigned)DIMK + (unsigned)(k0 + c0);
#pragma unroll
  for (int i = 0; i < 2; ++i)  // +64 rows per step
    async_b128(l0 + 3072u * i /*64*48*/, g0 + 131072u * i /*64*DIMK*/, src);
}

// 8 k-steps over one staged chunk: 64 ds_load_b64 feeding 128 WMMAs.
__device__ __forceinline__ void mma_chunk_f32(const float (*As)[36],
                                              const float (*Bs)[36], int wm,
                                              int wn, int lane,
                                              v8f acc[4][4]) {
  int ar = wm * 64 + (lane & 15);
  int br = wn * 64 + (lane & 15);
  int kk = (lane >> 4) << 1;
#pragma unroll
  for (int k = 0; k < KC; k += 4) {
    v2f af[4], bf[4];
#pragma unroll
    for (int t = 0; t < 4; ++t) {
      af[t] = *(const v2f*)&As[ar + t * 16][k + kk];
      bf[t] = *(const v2f*)&Bs[br + t * 16][k + kk];
    }
#pragma unroll
    for (int mi = 0; mi < 4; ++mi)
#pragma unroll
      for (int ni = 0; ni < 4; ++ni)
        acc[mi][ni] = wmma_f32x4(af[mi], bf[ni], acc[mi][ni]);
  }
}

__device__ __forceinline__ void mma_chunk_u8(const unsigned char (*As)[48],
                                             const float (*Bs)[36], int wm,
                                             int wn, int lane, v8f acc[4][4]) {
  int ar = wm * 64 + (lane & 15);
  int br = wn * 64 + (lane & 15);
  int kk = (lane >> 4) << 1;
#pragma unroll
  for (int k = 0; k < KC; k += 4) {
    v2f af[4], bf[4];
#pragma unroll
    for (int t = 0; t < 4; ++t) {
      const unsigned char* q = &As[ar + t * 16][k + kk];
      v2f a;
      a[0] = (float)q[0];
      a[1] = (float)q[1];
      af[t] = a;
      bf[t] = *(const v2f*)&Bs[br + t * 16][k + kk];
    }
#pragma unroll
    for (int mi = 0; mi < 4; ++mi)
#pragma unroll
      for (int ni = 0; ni < 4; ++ni)
        acc[mi][ni] = wmma_f32x4(af[mi], bf[ni], acc[mi][ni]);
  }
}

// Double-buffered main loops (block tile 128x128, 4 waves as 2x2).
__device__ __forceinline__ void gemm_main_f32(const float* __restrict__ Xg,
                                              const float* __restrict__ Wg,
                                              int tileM, int tileN, TileF32* As,
                                              TileF32* Bs, v8f acc[4][4]) {
  const int t = threadIdx.x;
  const int lane = t & 31;
  const int wave = t >> 5;
  const int wm = wave >> 1, wn = wave & 1;
  stage_f32(As[0], Xg, tileM, 0, t);
  stage_f32(Bs[0], Wg, tileN, 0, t);
  for (int c = 0; c < NCHUNK; ++c) {
    if (c + 1 < NCHUNK) {
      int nb = (c + 1) & 1;
      stage_f32(As[nb], Xg, tileM, (c + 1) * KC, t);
      stage_f32(Bs[nb], Wg, tileN, (c + 1) * KC, t);
      wait_async<16>();  // 16 in flight for c+1 => chunk c resident
    } else {
      wait_async<0>();
    }
    __syncthreads();
    mma_chunk_f32(As[c & 1], Bs[c & 1], wm, wn, lane, acc);
    __syncthreads();
  }
}

__device__ __forceinline__ void gemm_main_u8(const unsigned char* __restrict__ Sg,
                                             const float* __restrict__ Wg,
                                             int tileM, int tileN, TileU8* As,
                                             TileF32* Bs, v8f acc[4][4]) {
  const int t = threadIdx.x;
  const int lane = t & 31;
  const int wave = t >> 5;
  const int wm = wave >> 1, wn = wave & 1;
  stage_u8(As[0], Sg, tileM, 0, t);
  stage_f32(Bs[0], Wg, tileN, 0, t);
  for (int c = 0; c < NCHUNK; ++c) {
    if (c + 1 < NCHUNK) {
      int nb = (c + 1) & 1;
      stage_u8(As[nb], Sg, tileM, (c + 1) * KC, t);
      stage_f32(Bs[nb], Wg, tileN, (c + 1) * KC, t);
      wait_async<10>();  // 2 + 8 in flight for c+1
    } else {
      wait_async<0>();
    }
    __syncthreads();
    mma_chunk_u8(As[c & 1], Bs[c & 1], wm, wn, lane, acc);
    __syncthreads();
  }
}

// ---------------------------------------------------------------------------
// Small prologue kernels
// ---------------------------------------------------------------------------
__global__ void ucl_pooled_kernel(const float* __restrict__ x,
                                  float* __restrict__ pooled) {
  int g = blockIdx.x * blockDim.x + threadIdx.x;
  if (g >= NBATCH * DIMK) return;
  int b = g >> 11;
  int d = g & (DIMK - 1);
  const float* p = x + (size_t)b * SEQ * DIMK + d;
  float s = 0.f;
  for (int i = 0; i < SEQ; ++i) s += p[(size_t)i * DIMK];
  pooled[g] = s * (1.0f / SEQ);
}

__global__ void ucl_router_kernel(const float* __restrict__ pooled,
                                  const float* __restrict__ rw,
                                  const float* __restrict__ rb,
                                  float* __restrict__ coef) {
  __shared__ float logits[NBATCH * 3];
  int t = threadIdx.x;
  if (t < NBATCH * 3) {
    int b = t / 3, j = t % 3;
    float s = 0.f;
    const float* pp = pooled + b * DIMK;
    const float* wp = rw + j * DIMK;
    for (int d = 0; d < DIMK; ++d) s += pp[d] * wp[d];
    logits[t] = s + rb[j];
  }
  __syncthreads();
  if (t == 0) {
    float pw0[NBATCH], pw1[NBATCH];
    for (int b = 0; b < NBATCH; ++b) {
      float l0 = logits[b * 3 + 0], l1 = logits[b * 3 + 1],
            l2 = logits[b * 3 + 2];
      float mx = fmaxf(l0, fmaxf(l1, l2));
      float e0 = expf(l0 - mx), e1 = expf(l1 - mx), e2 = expf(l2 - mx);
      float inv = 1.f / (e0 + e1 + e2);
      pw0[b] = e0 * inv;
      pw1[b] = e1 * inv;
    }
    float m0 = 0.25f * (pw0[0] + pw0[1] + pw0[2] + pw0[3]);
    float m1 = 0.25f * (pw1[0] + pw1[1] + pw1[2] + pw1[3]);
    for (int b = 0; b < NBATCH; ++b) {
      coef[b] = (m0 > 0.3f) ? pw0[b] : 0.f;
      coef[NBATCH + b] = (m1 > 0.3f) ? pw1[b] : 0.f;
    }
  }
}

__global__ void ucl_c2_kernel(const float* __restrict__ cb,
                              float* __restrict__ c2) {
  int k = blockIdx.x * blockDim.x + threadIdx.x;
  if (k >= NSYM) return;
  const float* p = cb + (size_t)k * DIMK;
  float s = 0.f;
  for (int d = 0; d < DIMK; ++d) s += p[d] * p[d];
  c2[k] = s;
}

__global__ void ucl_init_argbuf(unsigned long long* __restrict__ argbuf) {
  int g = blockIdx.x * blockDim.x + threadIdx.x;
  if (g < MTOT) argbuf[g] = ~0ull;
}

// ---------------------------------------------------------------------------
// GEMM kernels.  C layout of a 16x16 WMMA tile: vgpr i, lanes 0-15 ->
// (M=i, N=lane); lanes 16-31 -> (M=8+i, N=lane-16).
// ---------------------------------------------------------------------------
__global__ __launch_bounds__(128) void ucl_gemm_neural(
    const float* __restrict__ X, const float* __restrict__ W,
    const float* __restrict__ bias, float* __restrict__ out) {
  __shared__ struct { TileF32 A[2]; TileF32 B[2]; } sm;
  int lane = threadIdx.x & 31, wave = threadIdx.x >> 5;
  int tileM = blockIdx.y * 128, tileN = blockIdx.x * 128;
  v8f acc[4][4] = {};
  gemm_main_f32(X, W, tileM, tileN, sm.A, sm.B, acc);
  int rowB = tileM + (wave >> 1) * 64 + (lane >> 4) * 8;
  int colB = tileN + (wave & 1) * 64 + (lane & 15);
#pragma unroll
  for (int mi = 0; mi < 4; ++mi)
#pragma unroll
    for (int ni = 0; ni < 4; ++ni)
#pragma unroll
      for (int i = 0; i < 8; ++i) {
        int row = rowB + mi * 16 + i;
        int col = colB + ni * 16;
        out[(size_t)row * DIMK + col] = acc[mi][ni][i] + bias[col];
      }
}

__global__ __launch_bounds__(128) void ucl_gemm_enc(
    const float* __restrict__ X, const float* __restrict__ W,
    const float* __restrict__ bias, const float* __restrict__ thresh,
    unsigned char* __restrict__ spikes) {
  __shared__ struct { TileF32 A[2]; TileF32 B[2]; } sm;
  int lane = threadIdx.x & 31, wave = threadIdx.x >> 5;
  int tileM = blockIdx.y * 128, tileN = blockIdx.x * 128;
  float t = *thresh;
  float lthr = logf(t / (1.f - t));  // sigmoid(v) > t  <=>  v > logit(t)
  v8f acc[4][4] = {};
  gemm_main_f32(X, W, tileM, tileN, sm.A, sm.B, acc);
  int rowB = tileM + (wave >> 1) * 64 + (lane >> 4) * 8;
  int colB = tileN + (wave & 1) * 64 + (lane & 15);
#pragma unroll
  for (int mi = 0; mi < 4; ++mi)
#pragma unroll
    for (int ni = 0; ni < 4; ++ni)
#pragma unroll
      for (int i = 0; i < 8; ++i) {
        int row = rowB + mi * 16 + i;
        int col = colB + ni * 16;
        float v = acc[mi][ni][i] + bias[col];
        spikes[(size_t)row * DIMK + col] =
            (v > lthr) ? (unsigned char)1 : (unsigned char)0;
      }
}

__device__ __forceinline__ unsigned long long ucl_shfl_xor_u64(
    unsigned long long v, int m) {
  unsigned int lo = (unsigned int)v;
  unsigned int hi = (unsigned int)(v >> 32);
  lo = __shfl_xor(lo, m, 32);
  hi = __shfl_xor(hi, m, 32);
  return ((unsigned long long)hi << 32) | lo;
}

__global__ __launch_bounds__(128) void ucl_gemm_code(
    const float* __restrict__ X, const float* __restrict__ cb,
    const float* __restrict__ c2, unsigned long long* __restrict__ argbuf) {
  __shared__ struct { TileF32 A[2]; TileF32 B[2]; } sm;
  int lane = threadIdx.x & 31, wave = threadIdx.x >> 5;
  int tileM = blockIdx.y * 128, tileN = blockIdx.x * 128;
  v8f acc[4][4] = {};
  gemm_main_f32(X, cb, tileM, tileN, sm.A, sm.B, acc);
  int rowB = tileM + (wave >> 1) * 64 + (lane >> 4) * 8;
  int colB = tileN + (wave & 1) * 64 + (lane & 15);
  int cl = lane & 15;
#pragma unroll
  for (int mi = 0; mi < 4; ++mi)
#pragma unroll
    for (int i = 0; i < 8; ++i) {
      unsigned long long best = ~0ull;  // local argmin over 4 N-subtiles
#pragma unroll
      for (int ni = 0; ni < 4; ++ni) {
        int col = colB + ni * 16;
        float score = c2[col] - 2.0f * acc[mi][ni][i];  // x^2 row-constant
        unsigned int fb = __float_as_uint(score);
        unsigned int key = (fb & 0x80000000u) ? ~fb : (fb | 0x80000000u);
        unsigned long long pk =
            ((unsigned long long)key << 32) | (unsigned int)col;
        if (pk < best) best = pk;
      }
#pragma unroll
      for (int m = 1; m < 16; m <<= 1) {  // same-row 16-lane reduction
        unsigned long long o = ucl_shfl_xor_u64(best, m);
        if (o < best) best = o;
      }
      if (cl == 0) atomicMin(argbuf + rowB + mi * 16 + i, best);
    }
}

__global__ __launch_bounds__(128) void ucl_gemm_dec(
    const unsigned char* __restrict__ spikes, const float* __restrict__ W,
    const float* __restrict__ bias, const float* __restrict__ coef,
    const unsigned long long* __restrict__ argbuf,
    const float* __restrict__ cb, float* __restrict__ out) {
  __shared__ struct { TileU8 A[2]; TileF32 B[2]; } sm;
  int lane = threadIdx.x & 31, wave = threadIdx.x >> 5;
  int tileM = blockIdx.y * 128, tileN = blockIdx.x * 128;
  v8f acc[4][4] = {};
  gemm_main_u8(spikes, W, tileM, tileN, sm.A, sm.B, acc);
  int rowB = tileM + (wave >> 1) * 64 + (lane >> 4) * 8;
  int colB = tileN + (wave & 1) * 64 + (lane & 15);
  int b = tileM >> 12;  // SEQ=4096 per batch; 128-row tiles never straddle
  float a0 = coef[b];
  float a1 = coef[NBATCH + b];
#pragma unroll
  for (int mi = 0; mi < 4; ++mi)
#pragma unroll
    for (int i = 0; i < 8; ++i) {
      int row = rowB + mi * 16 + i;
      int idx = (int)(argbuf[row] & 0xFFFFFFFFu);
      const float* crow = cb + (size_t)idx * DIMK;
#pragma unroll
      for (int ni = 0; ni < 4; ++ni) {
        int col = colB + ni * 16;
        size_t o = (size_t)row * DIMK + col;
        float add = a0 * (acc[mi][ni][i] + bias[col]) + a1 * crow[col];
        out[o] = out[o] + add;
      }
    }
}

// ---------------------------------------------------------------------------
// Workspace layout (bytes):
//   [0,      32768)  pooled   f32[4*2048]
//   [32768,  32800)  coef     f32[8]  (a0[4], a1[4])
//   [36864,  40960)  c2       f32[1024]
//   [65536, 196608)  argbuf   u64[16384]
//   [262144, 262144+32MiB)  spikes u8[16384*2048]
// ---------------------------------------------------------------------------
extern "C" void kernel_launch(void* const* d_in, const int* in_sizes, int n_in,
                              void* d_out, int out_size, void* d_ws,
                              size_t ws_size, hipStream_t stream) {
  (void)in_sizes; (void)n_in; (void)out_size; (void)ws_size;
  const float* x      = (const float*)d_in[0];
  const float* nw     = (const float*)d_in[1];
  const float* nb     = (const float*)d_in[2];
  const float* ew     = (const float*)d_in[3];
  const float* eb     = (const float*)d_in[4];
  const float* thresh = (const float*)d_in[5];
  const float* dw     = (const float*)d_in[6];
  const float* db     = (const float*)d_in[7];
  const float* cbk    = (const float*)d_in[8];
  const float* rw     = (const float*)d_in[9];
  const float* rb     = (const float*)d_in[10];
  float* out = (float*)d_out;

  char* ws = (char*)d_ws;
  float* pooled = (float*)(ws + 0);
  float* coef   = (float*)(ws + 32768);
  float* c2     = (float*)(ws + 36864);
  unsigned long long* argbuf = (unsigned long long*)(ws + 65536);
  unsigned char* spikes = (unsigned char*)(ws + 262144);

  ucl_pooled_kernel<<<(NBATCH * DIMK) / 256, 256, 0, stream>>>(x, pooled);
  ucl_c2_kernel<<<NSYM / 256, 256, 0, stream>>>(cbk, c2);
  ucl_router_kernel<<<1, 32, 0, stream>>>(pooled, rw, rb, coef);
  ucl_init_argbuf<<<MTOT / 256, 256, 0, stream>>>(argbuf);

  const int MB = MTOT / 128;  // 128 M-blocks
  ucl_gemm_neural<<<dim3(DIMK / 128, MB), 128, 0, stream>>>(x, nw, nb, out);
  ucl_gemm_enc<<<dim3(DIMK / 128, MB), 128, 0, stream>>>(x, ew, eb, thresh,
                                                         spikes);
  ucl_gemm_code<<<dim3(NSYM / 128, MB), 128, 0, stream>>>(x, cbk, c2, argbuf);
  ucl_gemm_dec<<<dim3(DIMK / 128, MB), 128, 0, stream>>>(spikes, dw, db, coef,
                                                         argbuf, cbk, out);
}